// GNNLayer_19215683682942
// MI455X (gfx1250) — compile-verified
//
#include <hip/hip_runtime.h>
#include <hip/hip_bf16.h>

// ---------------------------------------------------------------------------
// Equivariant GNN layer for MI455X (gfx1250, wave32).
// Every fctp bilinear contraction is a GEMM [16 rows] x [K path-features] x
// [N out channels] on V_WMMA_F32_16X16X4_F32 (full fp32 on matrix cores).
// K-loops are split per feature region (branch-free bodies); weight tables are
// pre-baked in per-WMMA *fragment order* so each lane does one coalesced b64
// load per step; path normalization constants are folded into the tables.
// Gate activations are applied in-fragment (silu on scalar D-frags, sigmoid
// gates kept in registers between the scalar and vector GEMM phases).
// ---------------------------------------------------------------------------

typedef float v2f __attribute__((ext_vector_type(2)));
typedef float v8f __attribute__((ext_vector_type(8)));

#define WMMA_F32X4(a, b, c) \
  __builtin_amdgcn_wmma_f32_16x16x4_f32(false, (a), false, (b), 0, (c), false, false)
#define ZERO8 {0.f, 0.f, 0.f, 0.f, 0.f, 0.f, 0.f, 0.f}

__device__ __forceinline__ float dot3(const float* a, const float* b) {
  return a[0] * b[0] + a[1] * b[1] + a[2] * b[2];
}
__device__ __forceinline__ float siluf(float x) { return x / (1.f + __expf(-x)); }
__device__ __forceinline__ float sigmf(float x) { return 1.f / (1.f + __expf(-x)); }

// --------------------------- path normalization ----------------------------
constexpr float KI2 = 0.70710678118654752f;   // 1/sqrt(2)
constexpr float KI3 = 0.57735026918962576f;   // 1/sqrt(3)
constexpr float KR128 = 0.088388347648318447f; // 1/sqrt(128)
constexpr float S_A = KI2 / 16.f;             // ss, stages 1/3
constexpr float S_B = KI2 * KI3 / 8.f;        // vvs, all stages
constexpr float S_C = KI3 * KR128;            // sv/vs, fan-in 128
constexpr float S_D = KI3 * KI2 / 8.f;        // vvv, all stages
constexpr float S_E = KI2 * KR128;            // ss stage 2
constexpr float S_F = KI3 / 8.f;              // vs stage 2, fan-in 64

// ---------------- scaled weight lookup (logical [K][N] view) ---------------
__device__ __forceinline__ float wgtS1(int k, int n, const float* Wss, const float* Wvvs) {
  if (k < 256) { int i = k >> 4, j = k & 15; return Wss[(i * 16 + j) * 16 + n] * S_A; }
  int u = k - 256, i = u >> 3, j = u & 7;
  return Wvvs[(i * 8 + j) * 16 + n] * S_B;
}
__device__ __forceinline__ float wgtV1(int k, int n, const float* Wsv, const float* Wvs,
                                       const float* Wvvv) {
  if (n >= 8) return 0.f;
  if (k < 128) { int i = k >> 3, j = k & 7; return Wsv[(i * 8 + j) * 8 + n] * S_C; }
  if (k < 256) { int u = k - 128, i = u >> 4, j = u & 15; return Wvs[(i * 16 + j) * 8 + n] * S_C; }
  int u = k - 256, i = u >> 3, j = u & 7;
  return Wvvv[(i * 8 + j) * 8 + n] * S_D;
}
__device__ __forceinline__ float wgtS2(int k, int n, const float* Wss, const float* Wvvs) {
  if (k < 128) { int i = k >> 3, j = k & 7; return Wss[(i * 8 + j) * 48 + n] * S_E; }
  int u = k - 128, i = u >> 3, j = u & 7;
  return Wvvs[(i * 8 + j) * 48 + n] * S_B;
}
__device__ __forceinline__ float wgtV2(int k, int n, const float* Wsv, const float* Wvs,
                                       const float* Wvvv) {
  if (k < 128) { int i = k >> 3, j = k & 7; return Wsv[(i * 8 + j) * 16 + n] * S_C; }
  if (k < 192) { int u = k - 128, i = u >> 3, j = u & 7; return Wvs[(i * 8 + j) * 16 + n] * S_F; }
  int u = k - 192, i = u >> 3, j = u & 7;
  return Wvvv[(i * 8 + j) * 16 + n] * S_D;
}
__device__ __forceinline__ float wgtS3(int k, int n, const float* Wss, const float* Wvvs) {
  if (k < 256) { int i = k >> 4, j = k & 15; return Wss[(i * 16 + j) * 48 + n] * S_A; }
  int u = k - 256, i = u >> 3, j = u & 7;
  return Wvvs[(i * 8 + j) * 48 + n] * S_B;
}
__device__ __forceinline__ float wgtV3(int k, int n, const float* Wsv, const float* Wvs,
                                       const float* Wvvv) {
  if (k < 128) { int i = k >> 3, j = k & 7; return Wsv[(i * 8 + j) * 16 + n] * S_C; }
  if (k < 256) { int u = k - 128, i = u >> 4, j = u & 15; return Wvs[(i * 16 + j) * 16 + n] * S_C; }
  int u = k - 256, i = u >> 3, j = u & 7;
  return Wvvv[(i * 8 + j) * 16 + n] * S_D;
}

// ---------------------------------------------------------------------------
// Weight prep: emit tables in per-WMMA B-fragment order.
//   WB[step][lane][2]  where step = k0/4 (within an N-tile),
//   lane < 16 holds K = {k0, k0+1}, lane >= 16 holds K = {k0+2, k0+3},
//   column n = lane & 15.  One wave reads 256B contiguous per step.
// ---------------------------------------------------------------------------
__global__ void prep_kernel(
    const float* __restrict__ W1ss, const float* __restrict__ W1sv,
    const float* __restrict__ W1vs, const float* __restrict__ W1vvs,
    const float* __restrict__ W1vvv,
    const float* __restrict__ W2ss, const float* __restrict__ W2sv,
    const float* __restrict__ W2vs, const float* __restrict__ W2vvs,
    const float* __restrict__ W2vvv,
    const float* __restrict__ W3ss, const float* __restrict__ W3sv,
    const float* __restrict__ W3vs, const float* __restrict__ W3vvs,
    const float* __restrict__ W3vvv,
    float* __restrict__ WS1f, float* __restrict__ WV1f,
    float* __restrict__ WS2f, float* __restrict__ WV2f,
    float* __restrict__ WS3f, float* __restrict__ WV3f) {
  const int tid = blockIdx.x * blockDim.x + threadIdx.x;
  const int stride = gridDim.x * blockDim.x;

  for (int t = tid; t < 5120; t += stride) {   // WS1f: 80 steps
    int step = t >> 6, l = (t >> 1) & 31, jj = t & 1;
    int ka = step * 4 + ((l >> 4) << 1) + jj, n = l & 15;
    WS1f[t] = wgtS1(ka, n, W1ss, W1vvs);
  }
  for (int t = tid; t < 5120; t += stride) {   // WV1f: 80 steps
    int step = t >> 6, l = (t >> 1) & 31, jj = t & 1;
    int ka = step * 4 + ((l >> 4) << 1) + jj, n = l & 15;
    WV1f[t] = wgtV1(ka, n, W1sv, W1vs, W1vvv);
  }
  for (int t = tid; t < 9216; t += stride) {   // WS2f: 3 N-tiles x 48 steps
    int nt = t / 3072, r = t % 3072;
    int step = r >> 6, l = (r >> 1) & 31, jj = t & 1;
    int ka = step * 4 + ((l >> 4) << 1) + jj, n = nt * 16 + (l & 15);
    WS2f[t] = wgtS2(ka, n, W2ss, W2vvs);
  }
  for (int t = tid; t < 4096; t += stride) {   // WV2f: 64 steps
    int step = t >> 6, l = (t >> 1) & 31, jj = t & 1;
    int ka = step * 4 + ((l >> 4) << 1) + jj, n = l & 15;
    WV2f[t] = wgtV2(ka, n, W2sv, W2vs, W2vvv);
  }
  for (int t = tid; t < 15360; t += stride) {  // WS3f: 3 N-tiles x 80 steps
    int nt = t / 5120, r = t % 5120;
    int step = r >> 6, l = (r >> 1) & 31, jj = t & 1;
    int ka = step * 4 + ((l >> 4) << 1) + jj, n = nt * 16 + (l & 15);
    WS3f[t] = wgtS3(ka, n, W3ss, W3vvs);
  }
  for (int t = tid; t < 5120; t += stride) {   // WV3f: 80 steps
    int step = t >> 6, l = (t >> 1) & 31, jj = t & 1;
    int ka = step * 4 + ((l >> 4) << 1) + jj, n = l & 15;
    WV3f[t] = wgtV3(ka, n, W3sv, W3vs, W3vvv);
  }
}

__device__ __forceinline__ v2f ldb(const float* base, int k0) {
  return *(const v2f*)(base + (k0 << 4));   // (k0/4)*64 floats
}

// ---------------------------------------------------------------------------
// Message kernel: 4 waves/block, 16 edges per wave.
// ---------------------------------------------------------------------------
__global__ __launch_bounds__(128) void msg_kernel(
    const float* __restrict__ node_s, const float* __restrict__ node_v,
    const float* __restrict__ edge_s, const float* __restrict__ edge_v,
    const int* __restrict__ edge_index,
    const float* __restrict__ WS1f, const float* __restrict__ WV1f,
    const float* __restrict__ WS2f, const float* __restrict__ WV2f,
    const float* __restrict__ Lms, const float* __restrict__ Lmv,
    float* __restrict__ agg, int nE) {
  __shared__ float lds[4][3712];
  __shared__ int ridx[4][16], cidx[4][16];
  const int wave = threadIdx.x >> 5, lane = threadIdx.x & 31;
  float* Lb  = lds[wave];
  float* s1  = Lb;            // [16][16]
  float* s2  = s1 + 256;      // [16][16]
  float* v1  = s2 + 256;      // [16][24]
  float* v2  = v1 + 384;      // [16][24]
  float* es  = v2 + 384;      // [16][8]
  float* ev  = es + 128;      // [16][24]
  float* so1 = ev + 384;      // [16][16]
  float* vo1 = so1 + 256;     // [16][24]
  float* hact = vo1 + 384;    // [16][32]  silu-activated hidden scalars
  float* hvg  = hact + 512;   // [16][48]  gated vectors [ch*3+c]

  const int e0 = (blockIdx.x * 4 + wave) * 16;
  if (lane < 16) {
    int e = e0 + lane;
    bool ok = e < nE;
    ridx[wave][lane] = ok ? edge_index[e] : 0;
    cidx[wave][lane] = ok ? edge_index[nE + e] : -1;
  }
  __syncthreads();
  for (int t = lane; t < 256; t += 32) {
    int r = t >> 4, ch = t & 15;
    int ri = ridx[wave][r], ci = cidx[wave][r];
    bool ok = ci >= 0;
    s1[t] = ok ? node_s[ri * 16 + ch] : 0.f;
    s2[t] = ok ? node_s[ci * 16 + ch] : 0.f;
  }
  for (int t = lane; t < 384; t += 32) {
    int r = t / 24, ch = t % 24;
    int ri = ridx[wave][r], ci = cidx[wave][r];
    bool ok = ci >= 0;
    v1[t] = ok ? node_v[ri * 24 + ch] : 0.f;
    v2[t] = ok ? node_v[ci * 24 + ch] : 0.f;
    ev[t] = (e0 + r < nE) ? edge_v[(e0 + r) * 24 + ch] : 0.f;
  }
  for (int t = lane; t < 128; t += 32) {
    int r = t >> 3, ch = t & 7;
    es[t] = (e0 + r < nE) ? edge_s[(e0 + r) * 8 + ch] : 0.f;
  }
  __syncthreads();

  const int mrow  = lane & 15;
  const int khi   = (lane >> 4) << 1;
  const int mbase = (lane >> 4) << 3;
  const float* s1r = s1 + mrow * 16;
  const float* s2r = s2 + mrow * 16;
  const float* v1r = v1 + mrow * 24;
  const float* v2r = v2 + mrow * 24;
  const float* esr = es + mrow * 8;
  const float* evr = ev + mrow * 24;
  const float* so1r = so1 + mrow * 16;
  const float* vo1r = vo1 + mrow * 24;
  const float* BS1 = WS1f + lane * 2;
  const float* BV1 = WV1f + lane * 2;
  const float* BS2 = WS2f + lane * 2;
  const float* BV2 = WV2f + lane * 2;

  // ---- stage 1 scalar: [16x320] x [320x16] ----
  {
    v8f acc = ZERO8;
#pragma unroll 4
    for (int k0 = 0; k0 < 256; k0 += 4) {        // s16 x s16
      int ka = k0 + khi, i = ka >> 4, j = ka & 15;
      float sv = s1r[i];
      v2f a; a.x = sv * s2r[j]; a.y = sv * s2r[j + 1];
      acc = WMMA_F32X4(a, ldb(BS1, k0), acc);
    }
#pragma unroll 4
    for (int k0 = 256; k0 < 320; k0 += 4) {      // dot(v8, v8)
      int u = k0 - 256 + khi, i = u >> 3, j = u & 7;
      const float* va = v1r + i * 3;
      v2f a; a.x = dot3(va, v2r + j * 3); a.y = dot3(va, v2r + (j + 1) * 3);
      acc = WMMA_F32X4(a, ldb(BS1, k0), acc);
    }
#pragma unroll
    for (int v = 0; v < 8; ++v) so1[(v + mbase) * 16 + mrow] = acc[v];
  }
  // ---- stage 1 vector: 3 comps, [16x320] x [320x16] ----
  for (int c = 0; c < 3; ++c) {
    int c1 = (c == 2) ? 0 : c + 1, c2 = (c == 0) ? 2 : c - 1;
    v8f acc = ZERO8;
#pragma unroll 4
    for (int k0 = 0; k0 < 128; k0 += 4) {        // s16 x v8c
      int ka = k0 + khi, i = ka >> 3, j = ka & 7;
      float sv = s1r[i];
      v2f a; a.x = sv * v2r[j * 3 + c]; a.y = sv * v2r[(j + 1) * 3 + c];
      acc = WMMA_F32X4(a, ldb(BV1, k0), acc);
    }
#pragma unroll 4
    for (int k0 = 128; k0 < 256; k0 += 4) {      // v8c x s16
      int u = k0 - 128 + khi, i = u >> 4, j = u & 15;
      float vv = v1r[i * 3 + c];
      v2f a; a.x = vv * s2r[j]; a.y = vv * s2r[j + 1];
      acc = WMMA_F32X4(a, ldb(BV1, k0), acc);
    }
#pragma unroll 4
    for (int k0 = 256; k0 < 320; k0 += 4) {      // cross(v8, v8)_c
      int u = k0 - 256 + khi, i = u >> 3, j = u & 7;
      float a1 = v1r[i * 3 + c1], a2 = v1r[i * 3 + c2];
      v2f a;
      a.x = a1 * v2r[j * 3 + c2] - a2 * v2r[j * 3 + c1];
      a.y = a1 * v2r[(j + 1) * 3 + c2] - a2 * v2r[(j + 1) * 3 + c1];
      acc = WMMA_F32X4(a, ldb(BV1, k0), acc);
    }
    if (mrow < 8) {
#pragma unroll
      for (int v = 0; v < 8; ++v) vo1[(v + mbase) * 24 + mrow * 3 + c] = acc[v];
    }
  }
  __syncthreads();
  // ---- stage 2 scalar: [16x192] x [192x48], 3 N-tiles share the A frag ----
  float gte[8];   // sigmoid gates, kept in registers for the vector phase
  {
    v8f a0 = ZERO8, a1v = ZERO8, a2v = ZERO8;
#pragma unroll 4
    for (int k0 = 0; k0 < 128; k0 += 4) {        // so16 x es8
      int ka = k0 + khi, i = ka >> 3, j = ka & 7;
      float sv = so1r[i];
      v2f a; a.x = sv * esr[j]; a.y = sv * esr[j + 1];
      a0  = WMMA_F32X4(a, ldb(BS2, k0), a0);
      a1v = WMMA_F32X4(a, ldb(BS2 + 3072, k0), a1v);
      a2v = WMMA_F32X4(a, ldb(BS2 + 6144, k0), a2v);
    }
#pragma unroll 4
    for (int k0 = 128; k0 < 192; k0 += 4) {      // dot(vo8, ev8)
      int u = k0 - 128 + khi, i = u >> 3, j = u & 7;
      const float* va = vo1r + i * 3;
      v2f a; a.x = dot3(va, evr + j * 3); a.y = dot3(va, evr + (j + 1) * 3);
      a0  = WMMA_F32X4(a, ldb(BS2, k0), a0);
      a1v = WMMA_F32X4(a, ldb(BS2 + 3072, k0), a1v);
      a2v = WMMA_F32X4(a, ldb(BS2 + 6144, k0), a2v);
    }
#pragma unroll
    for (int v = 0; v < 8; ++v) {
      hact[(v + mbase) * 32 + mrow]      = siluf(a0[v]);   // ch 0..15
      hact[(v + mbase) * 32 + 16 + mrow] = siluf(a1v[v]);  // ch 16..31
      gte[v] = sigmf(a2v[v]);                              // gate ch = mrow
    }
  }
  __syncthreads();
  // ---- stage 2 vector: 3 comps, [16x256] x [256x16], gated in-register ----
  for (int c = 0; c < 3; ++c) {
    int c1 = (c == 2) ? 0 : c + 1, c2 = (c == 0) ? 2 : c - 1;
    v8f acc = ZERO8;
#pragma unroll 4
    for (int k0 = 0; k0 < 128; k0 += 4) {        // so16 x ev8c
      int ka = k0 + khi, i = ka >> 3, j = ka & 7;
      float sv = so1r[i];
      v2f a; a.x = sv * evr[j * 3 + c]; a.y = sv * evr[(j + 1) * 3 + c];
      acc = WMMA_F32X4(a, ldb(BV2, k0), acc);
    }
#pragma unroll 4
    for (int k0 = 128; k0 < 192; k0 += 4) {      // vo8c x es8
      int u = k0 - 128 + khi, i = u >> 3, j = u & 7;
      float vv = vo1r[i * 3 + c];
      v2f a; a.x = vv * esr[j]; a.y = vv * esr[j + 1];
      acc = WMMA_F32X4(a, ldb(BV2, k0), acc);
    }
#pragma unroll 4
    for (int k0 = 192; k0 < 256; k0 += 4) {      // cross(vo8, ev8)_c
      int u = k0 - 192 + khi, i = u >> 3, j = u & 7;
      float a1 = vo1r[i * 3 + c1], a2 = vo1r[i * 3 + c2];
      v2f a;
      a.x = a1 * evr[j * 3 + c2] - a2 * evr[j * 3 + c1];
      a.y = a1 * evr[(j + 1) * 3 + c2] - a2 * evr[(j + 1) * 3 + c1];
      acc = WMMA_F32X4(a, ldb(BV2, k0), acc);
    }
#pragma unroll
    for (int v = 0; v < 8; ++v)
      hvg[(v + mbase) * 48 + mrow * 3 + c] = acc[v] * gte[v];
  }
  __syncthreads();
  // ---- Lm linear + atomic scatter-sum into agg[col] ----
  for (int idx = lane; idx < 640; idx += 32) {
    int e = idx / 40, o = idx % 40;
    int ci = cidx[wave][e];
    if (ci < 0) continue;
    float val = 0.f;
    if (o < 16) {
#pragma unroll
      for (int i = 0; i < 32; ++i) val += hact[e * 32 + i] * Lms[i * 16 + o];
      val *= 0.17677669529663687f;   // 1/sqrt(32)
    } else {
      int o2 = o - 16, kk = o2 / 3, c = o2 % 3;
#pragma unroll
      for (int i = 0; i < 16; ++i) val += hvg[e * 48 + i * 3 + c] * Lmv[i * 8 + kk];
      val *= 0.25f;                  // 1/sqrt(16)
    }
    atomicAdd(&agg[ci * 40 + o], val);
  }
}

// ---------------------------------------------------------------------------
// Update kernel: 4 waves/block, 16 nodes per wave.
// ---------------------------------------------------------------------------
__global__ __launch_bounds__(128) void upd_kernel(
    const float* __restrict__ node_s, const float* __restrict__ node_v,
    const float* __restrict__ agg,
    const float* __restrict__ WS3f, const float* __restrict__ WV3f,
    const float* __restrict__ Lus, const float* __restrict__ Luv,
    float* __restrict__ out, int nN) {
  __shared__ float lds[4][2560];
  const int wave = threadIdx.x >> 5, lane = threadIdx.x & 31;
  float* Lb   = lds[wave];
  float* ns   = Lb;           // [16][16]
  float* nv   = ns + 256;     // [16][24]
  float* asd  = nv + 384;     // [16][16]
  float* av   = asd + 256;    // [16][24]
  float* hact = av + 384;     // [16][32]
  float* hvg  = hact + 512;   // [16][48]
  const int n0 = (blockIdx.x * 4 + wave) * 16;
  for (int t = lane; t < 256; t += 32) {
    int r = t >> 4, ch = t & 15;
    int n = n0 + r; bool ok = n < nN;
    ns[t]  = ok ? node_s[n * 16 + ch] : 0.f;
    asd[t] = ok ? agg[n * 40 + ch] : 0.f;
  }
  for (int t = lane; t < 384; t += 32) {
    int r = t / 24, ch = t % 24;
    int n = n0 + r; bool ok = n < nN;
    nv[t] = ok ? node_v[n * 24 + ch] : 0.f;
    av[t] = ok ? agg[n * 40 + 16 + ch] : 0.f;
  }
  __syncthreads();

  const int mrow  = lane & 15;
  const int khi   = (lane >> 4) << 1;
  const int mbase = (lane >> 4) << 3;
  const float* nsr = ns + mrow * 16;
  const float* asr = asd + mrow * 16;
  const float* nvr = nv + mrow * 24;
  const float* avr = av + mrow * 24;
  const float* BS3 = WS3f + lane * 2;
  const float* BV3 = WV3f + lane * 2;

  // ---- scalar: [16x320] x [320x48], 3 N-tiles share the A frag ----
  float gte[8];
  {
    v8f a0 = ZERO8, a1v = ZERO8, a2v = ZERO8;
#pragma unroll 4
    for (int k0 = 0; k0 < 256; k0 += 4) {        // s16 x a16
      int ka = k0 + khi, i = ka >> 4, j = ka & 15;
      float sv = nsr[i];
      v2f a; a.x = sv * asr[j]; a.y = sv * asr[j + 1];
      a0  = WMMA_F32X4(a, ldb(BS3, k0), a0);
      a1v = WMMA_F32X4(a, ldb(BS3 + 5120, k0), a1v);
      a2v = WMMA_F32X4(a, ldb(BS3 + 10240, k0), a2v);
    }
#pragma unroll 4
    for (int k0 = 256; k0 < 320; k0 += 4) {      // dot(nv8, av8)
      int u = k0 - 256 + khi, i = u >> 3, j = u & 7;
      const float* va = nvr + i * 3;
      v2f a; a.x = dot3(va, avr + j * 3); a.y = dot3(va, avr + (j + 1) * 3);
      a0  = WMMA_F32X4(a, ldb(BS3, k0), a0);
      a1v = WMMA_F32X4(a, ldb(BS3 + 5120, k0), a1v);
      a2v = WMMA_F32X4(a, ldb(BS3 + 10240, k0), a2v);
    }
#pragma unroll
    for (int v = 0; v < 8; ++v) {
      hact[(v + mbase) * 32 + mrow]      = siluf(a0[v]);
      hact[(v + mbase) * 32 + 16 + mrow] = siluf(a1v[v]);
      gte[v] = sigmf(a2v[v]);
    }
  }
  __syncthreads();
  // ---- vector: 3 comps, [16x320] x [320x16], gated in-register ----
  for (int c = 0; c < 3; ++c) {
    int c1 = (c == 2) ? 0 : c + 1, c2 = (c == 0) ? 2 : c - 1;
    v8f acc = ZERO8;
#pragma unroll 4
    for (int k0 = 0; k0 < 128; k0 += 4) {        // s16 x av8c
      int ka = k0 + khi, i = ka >> 3, j = ka & 7;
      float sv = nsr[i];
      v2f a; a.x = sv * avr[j * 3 + c]; a.y = sv * avr[(j + 1) * 3 + c];
      acc = WMMA_F32X4(a, ldb(BV3, k0), acc);
    }
#pragma unroll 4
    for (int k0 = 128; k0 < 256; k0 += 4) {      // nv8c x a16
      int u = k0 - 128 + khi, i = u >> 4, j = u & 15;
      float vv = nvr[i * 3 + c];
      v2f a; a.x = vv * asr[j]; a.y = vv * asr[j + 1];
      acc = WMMA_F32X4(a, ldb(BV3, k0), acc);
    }
#pragma unroll 4
    for (int k0 = 256; k0 < 320; k0 += 4) {      // cross(nv8, av8)_c
      int u = k0 - 256 + khi, i = u >> 3, j = u & 7;
      float a1 = nvr[i * 3 + c1], a2 = nvr[i * 3 + c2];
      v2f a;
      a.x = a1 * avr[j * 3 + c2] - a2 * avr[j * 3 + c1];
      a.y = a1 * avr[(j + 1) * 3 + c2] - a2 * avr[(j + 1) * 3 + c1];
      acc = WMMA_F32X4(a, ldb(BV3, k0), acc);
    }
#pragma unroll
    for (int v = 0; v < 8; ++v)
      hvg[(v + mbase) * 48 + mrow * 3 + c] = acc[v] * gte[v];
  }
  __syncthreads();
  // ---- Lu linear + store [N][40] ----
  for (int idx = lane; idx < 640; idx += 32) {
    int r = idx / 40, o = idx % 40;
    int n = n0 + r;
    if (n >= nN) continue;
    float val = 0.f;
    if (o < 16) {
#pragma unroll
      for (int i = 0; i < 32; ++i) val += hact[r * 32 + i] * Lus[i * 16 + o];
      val *= 0.17677669529663687f;
    } else {
      int o2 = o - 16, kk = o2 / 3, c = o2 % 3;
#pragma unroll
      for (int i = 0; i < 16; ++i) val += hvg[r * 48 + i * 3 + c] * Luv[i * 8 + kk];
      val *= 0.25f;
    }
    out[n * 40 + o] = val;
  }
}

extern "C" void kernel_launch(void* const* d_in, const int* in_sizes, int n_in,
                              void* d_out, int out_size, void* d_ws, size_t ws_size,
                              hipStream_t stream) {
  const float* node_s = (const float*)d_in[0];
  const float* node_v = (const float*)d_in[1];
  // d_in[2] = pos (unused by the reference math)
  const float* edge_s = (const float*)d_in[3];
  const float* edge_v = (const float*)d_in[4];
  const int* edge_index = (const int*)d_in[5];
  const float* W1ss = (const float*)d_in[6];
  const float* W1sv = (const float*)d_in[7];
  const float* W1vs = (const float*)d_in[8];
  const float* W1vvs = (const float*)d_in[9];
  const float* W1vvv = (const float*)d_in[10];
  const float* W2ss = (const float*)d_in[11];
  const float* W2sv = (const float*)d_in[12];
  const float* W2vs = (const float*)d_in[13];
  const float* W2vvs = (const float*)d_in[14];
  const float* W2vvv = (const float*)d_in[15];
  const float* Lms = (const float*)d_in[16];
  const float* Lmv = (const float*)d_in[17];
  const float* W3ss = (const float*)d_in[18];
  const float* W3sv = (const float*)d_in[19];
  const float* W3vs = (const float*)d_in[20];
  const float* W3vvs = (const float*)d_in[21];
  const float* W3vvv = (const float*)d_in[22];
  const float* Lus = (const float*)d_in[23];
  const float* Luv = (const float*)d_in[24];

  const int nN = in_sizes[0] / 16;   // node_s is (N,16)
  const int nE = in_sizes[3] / 8;    // edge_s is (E,8)

  float* ws = (float*)d_ws;
  float* WS1f = ws;               // 5120
  float* WV1f = ws + 5120;        // 5120
  float* WS2f = ws + 10240;       // 9216
  float* WV2f = ws + 19456;       // 4096
  float* WS3f = ws + 23552;       // 15360
  float* WV3f = ws + 38912;       // 5120
  float* agg  = ws + 44032;       // N*40

  hipMemsetAsync(agg, 0, (size_t)nN * 40 * sizeof(float), stream);
  prep_kernel<<<48, 256, 0, stream>>>(W1ss, W1sv, W1vs, W1vvs, W1vvv,
                                      W2ss, W2sv, W2vs, W2vvs, W2vvv,
                                      W3ss, W3sv, W3vs, W3vvs, W3vvv,
                                      WS1f, WV1f, WS2f, WV2f, WS3f, WV3f);
  const int mblocks = (nE + 63) / 64;
  msg_kernel<<<mblocks, 128, 0, stream>>>(node_s, node_v, edge_s, edge_v, edge_index,
                                          WS1f, WV1f, WS2f, WV2f, Lms, Lmv, agg, nE);
  const int ublocks = (nN + 63) / 64;
  upd_kernel<<<ublocks, 128, 0, stream>>>(node_s, node_v, agg, WS3f, WV3f,
                                          Lus, Luv, (float*)d_out, nN);
}